// SurrogateModel_18537078849575
// MI455X (gfx1250) — compile-verified
//
#include <hip/hip_runtime.h>

#define N_NODES 50000
#define N_EDGES 800000
#define IN_DIM  128
#define HID     128
#define OUT_DIM 40
#define OUT_PAD 48        // 40 padded to 3 x 16-col tiles

typedef __attribute__((ext_vector_type(2))) float v2f;
typedef __attribute__((ext_vector_type(8))) float v8f;

// ---------------- degree / normalization ----------------

__global__ __launch_bounds__(256) void k_deg_init(float* __restrict__ deg) {
    int n = blockIdx.x * blockDim.x + threadIdx.x;
    if (n < N_NODES) deg[n] = 1.0f;   // self-loop contributes 1 to every node
}

__global__ __launch_bounds__(256) void k_deg_accum(const int* __restrict__ dst,
                                                   float* __restrict__ deg) {
    int e = blockIdx.x * blockDim.x + threadIdx.x;
    if (e < N_EDGES) atomicAdd(&deg[dst[e]], 1.0f);
}

__global__ __launch_bounds__(256) void k_dinv(float* __restrict__ deg) {
    int n = blockIdx.x * blockDim.x + threadIdx.x;
    if (n < N_NODES) deg[n] = rsqrtf(deg[n]);   // deg >= 1 always
}

// zero-pad W2 [128 x 40] -> W2p [128 x 48]
__global__ __launch_bounds__(256) void k_pad_w2(const float* __restrict__ W2,
                                                float* __restrict__ W2p) {
    int i = blockIdx.x * blockDim.x + threadIdx.x;   // over 128*48
    if (i < IN_DIM * OUT_PAD) {
        int k = i / OUT_PAD, c = i % OUT_PAD;
        W2p[i] = (c < OUT_DIM) ? W2[k * OUT_DIM + c] : 0.0f;
    }
}

// ---------------- FP32 WMMA GEMM: C[M x NC] = A[M x 128] * B[128 x NC] ----------------
// NC compile-time (128 or 48). One block = one 16-row tile; wave w = col tile w.
// B is staged transposed in LDS so the B fragment is one contiguous ds_load_b64.
// Fragment layout per CDNA5 ISA 7.12.2 (lanes 0-15: K0/K1, lanes 16-31: K2/K3).

template <int NC>
__global__ __launch_bounds__(32 * (NC / 16)) void k_gemm16(
        const float* __restrict__ A, const float* __restrict__ B,
        float* __restrict__ C) {
    __shared__ float BT[NC * IN_DIM];        // BT[c*128 + k]
    constexpr int NTHREADS = 32 * (NC / 16);

    const int tid = threadIdx.x;
    // cooperative coalesced load + transpose of B (128 x NC)
    for (int i = tid; i < IN_DIM * NC; i += NTHREADS) {
        const int k = i / NC, c = i % NC;
        BT[c * IN_DIM + k] = B[i];
    }
    __syncthreads();

    const int wave = tid >> 5;               // col tile index
    const int lane = tid & 31;
    const int half = lane >> 4;              // 0 or 1
    const int m    = lane & 15;
    const int row0 = blockIdx.x * 16;
    const int col  = wave * 16 + m;

    const float* arow = A + (size_t)(row0 + m) * IN_DIM + (half << 1);
    const float* bcol = BT + col * IN_DIM + (half << 1);

    v8f acc = {};
#pragma unroll
    for (int kb = 0; kb < IN_DIM; kb += 4) {
        const v2f a = *(const v2f*)(arow + kb);   // global_load_b64
        const v2f b = *(const v2f*)(bcol + kb);   // ds_load_b64
        acc = __builtin_amdgcn_wmma_f32_16x16x4_f32(
            false, a, false, b, (short)0, acc, false, false);
    }

#pragma unroll
    for (int i = 0; i < 8; ++i) {
        const int r = row0 + i + (half << 3);
        C[(size_t)r * NC + col] = acc[i];
    }
}

// ---------------- message passing ----------------

// h[n,0:dim] = xw[n,0:dim] * dinv[n]^2  (self-loop term initializes accumulator)
__global__ void k_self(const float* __restrict__ xw, const float* __restrict__ dinv,
                       float* __restrict__ h, int dim, int ld) {
    const int n = blockIdx.x;
    const int f = threadIdx.x;
    if (f < dim) {
        const float di = dinv[n];
        h[(size_t)n * dim + f] = xw[(size_t)n * ld + f] * di * di;
    }
}

// layer-1 scatter: one wave per edge, float4 per lane (128 feats), 8 edges/block
__global__ __launch_bounds__(256) void k_scatter128(
        const int* __restrict__ src, const int* __restrict__ dst,
        const float* __restrict__ dinv, const float* __restrict__ xw,
        float* __restrict__ h) {
    const int e = blockIdx.x * 8 + (threadIdx.x >> 5);
    if (e >= N_EDGES) return;
    const int lane = threadIdx.x & 31;
    const int s = src[e], d = dst[e];
    const float w = dinv[s] * dinv[d];
    const float4 v = *(const float4*)(xw + (size_t)s * HID + lane * 4);
    float* hp = h + (size_t)d * HID + lane * 4;
    atomicAdd(hp + 0, v.x * w);
    atomicAdd(hp + 1, v.y * w);
    atomicAdd(hp + 2, v.z * w);
    atomicAdd(hp + 3, v.w * w);
}

// layer-2 scatter: one wave per edge over 40 feats (xw has padded stride 48)
__global__ __launch_bounds__(256) void k_scatter40(
        const int* __restrict__ src, const int* __restrict__ dst,
        const float* __restrict__ dinv, const float* __restrict__ xw,
        float* __restrict__ h) {
    const int e = blockIdx.x * 8 + (threadIdx.x >> 5);
    if (e >= N_EDGES) return;
    const int lane = threadIdx.x & 31;
    const int s = src[e], d = dst[e];
    const float w = dinv[s] * dinv[d];
    for (int f = lane; f < OUT_DIM; f += 32)
        atomicAdd(&h[(size_t)d * OUT_DIM + f], xw[(size_t)s * OUT_PAD + f] * w);
}

__global__ void k_bias_relu(float* __restrict__ h, const float* __restrict__ b, int dim) {
    const int n = blockIdx.x;
    const int f = threadIdx.x;
    if (f < dim) {
        const float v = h[(size_t)n * dim + f] + b[f];
        h[(size_t)n * dim + f] = fmaxf(v, 0.0f);
    }
}

// ---------------- softmax over 40 classes, one node per thread ----------------

__global__ __launch_bounds__(256) void k_softmax(const float* __restrict__ h,
                                                 const float* __restrict__ b,
                                                 float* __restrict__ out) {
    const int n = blockIdx.x * blockDim.x + threadIdx.x;
    if (n >= N_NODES) return;
    float v[OUT_DIM];
    float mx = -3.402823466e+38f;
#pragma unroll
    for (int i = 0; i < OUT_DIM; ++i) {
        v[i] = h[(size_t)n * OUT_DIM + i] + b[i];
        mx = fmaxf(mx, v[i]);
    }
    float s = 0.0f;
#pragma unroll
    for (int i = 0; i < OUT_DIM; ++i) {
        const float e = __expf(v[i] - mx);
        v[i] = e;
        s += e;
    }
    const float inv = 1.0f / s;
#pragma unroll
    for (int i = 0; i < OUT_DIM; ++i)
        out[(size_t)n * OUT_DIM + i] = v[i] * inv;
}

// ---------------- launch ----------------

extern "C" void kernel_launch(void* const* d_in, const int* in_sizes, int n_in,
                              void* d_out, int out_size, void* d_ws, size_t ws_size,
                              hipStream_t stream) {
    const float* x  = (const float*)d_in[0];
    const int*   ei = (const int*)d_in[1];      // [2, E] flat
    const int*   src = ei;                       // edge_index[0]
    const int*   dst = ei + N_EDGES;             // edge_index[1]
    const float* W1 = (const float*)d_in[2];
    const float* b1 = (const float*)d_in[3];
    const float* W2 = (const float*)d_in[4];
    const float* b2 = (const float*)d_in[5];
    float* out = (float*)d_out;

    // workspace carve-out (all f32)
    float* deg = (float*)d_ws;                              // N        (becomes dinv)
    float* xw1 = deg + N_NODES;                             // N*128
    float* h1  = xw1 + (size_t)N_NODES * HID;               // N*128
    float* xw2 = h1  + (size_t)N_NODES * HID;               // N*48 (padded)
    float* h2  = xw2 + (size_t)N_NODES * OUT_PAD;           // N*40
    float* W2p = h2  + (size_t)N_NODES * OUT_DIM;           // 128*48

    // normalization
    k_deg_init <<<(N_NODES + 255) / 256, 256, 0, stream>>>(deg);
    k_deg_accum<<<(N_EDGES + 255) / 256, 256, 0, stream>>>(dst, deg);
    k_dinv     <<<(N_NODES + 255) / 256, 256, 0, stream>>>(deg);   // deg -> dinv in place
    k_pad_w2   <<<(IN_DIM * OUT_PAD + 255) / 256, 256, 0, stream>>>(W2, W2p);

    // layer 1: xw1 = x @ W1 (WMMA, B^T in LDS), self + edge scatter, bias+relu
    k_gemm16<HID><<<N_NODES / 16, 32 * (HID / 16), 0, stream>>>(x, W1, xw1);
    k_self      <<<N_NODES, 128, 0, stream>>>(xw1, deg, h1, HID, HID);
    k_scatter128<<<(N_EDGES + 7) / 8, 256, 0, stream>>>(src, dst, deg, xw1, h1);
    k_bias_relu <<<N_NODES, 128, 0, stream>>>(h1, b1, HID);

    // layer 2: xw2 = h1 @ W2p (padded 48 cols), self + scatter, softmax
    k_gemm16<OUT_PAD><<<N_NODES / 16, 32 * (OUT_PAD / 16), 0, stream>>>(h1, W2p, xw2);
    k_self     <<<N_NODES, 64, 0, stream>>>(xw2, deg, h2, OUT_DIM, OUT_PAD);
    k_scatter40<<<(N_EDGES + 7) / 8, 256, 0, stream>>>(src, dst, deg, xw2, h2);
    k_softmax  <<<(N_NODES + 255) / 256, 256, 0, stream>>>(h2, b2, out);
}